// HashedCountFeatureBuilder_60593398612011
// MI455X (gfx1250) — compile-verified
//
#include <hip/hip_runtime.h>
#include <hip/hip_bf16.h>
#include <math.h>

// ---------------------------------------------------------------------------
// HashedCountFeatureBuilder for MI455X (gfx1250, wave32)
// fp32 WMMA (V_WMMA_F32_16X16X4_F32) for all GEMMs; LDS-atomic histogram.
// ---------------------------------------------------------------------------

#define SEQ        131072
#define MSETS      4096
#define SET_SIZE   1024
#define NUM_BINS   128
#define D_MODEL    1024
#define D_PHI      512
// (tok * 1315423911 + 13) % 128  ==  ((tok&127)*39 + 13) & 127  for tok >= 0
#define HASH_MULT_MOD  39
#define HASH_SEED      13

typedef __attribute__((ext_vector_type(2))) float v2f;
typedef __attribute__((ext_vector_type(8))) float v8f;

#define LDA_SH 68   // 64 + 4 pad: row-strided LDS reads hit 16 distinct banks

// ---------------------------------------------------------------------------
// Kernel 1: per-set hashed token histogram, normalized by set size.
// One block (256 threads) per set; 128-bin LDS histogram via ds_add_f32.
// ---------------------------------------------------------------------------
__global__ void counts_kernel(const int* __restrict__ token_ids,
                              const int* __restrict__ set_indices,
                              const int* __restrict__ set_sizes,
                              float* __restrict__ counts)
{
    __shared__ float bins[NUM_BINS];
    const int m = blockIdx.x;
    for (int b = threadIdx.x; b < NUM_BINS; b += 256) bins[b] = 0.0f;
    __syncthreads();

    const int* row = set_indices + (size_t)m * SET_SIZE;
#pragma unroll
    for (int i = 0; i < SET_SIZE / 256; ++i) {
        const int idx = row[threadIdx.x + (i << 8)];
        if (idx >= 0) {
            const int tok = token_ids[idx];
            const int bin = ((tok & (NUM_BINS - 1)) * HASH_MULT_MOD + HASH_SEED) & (NUM_BINS - 1);
            atomicAdd(&bins[bin], 1.0f);          // ds_add_f32
        }
    }
    __syncthreads();

    const float inv = 1.0f / fmaxf((float)set_sizes[m], 1.0f);
    for (int b = threadIdx.x; b < NUM_BINS; b += 256)
        counts[(size_t)m * NUM_BINS + b] = bins[b] * inv;
}

// ---------------------------------------------------------------------------
// Kernel 2: geom_bias output (third return value): BETA - GAMMA*|pi - pj|.
// One block per row i, float4 stores (64 MB total).
// ---------------------------------------------------------------------------
__global__ void geom_bias_kernel(const float* __restrict__ pos,
                                 float* __restrict__ out)
{
    const int i = blockIdx.x;
    const float pi = pos[i];
    float* row = out + (size_t)i * MSETS;
#pragma unroll
    for (int it = 0; it < MSETS / (256 * 4); ++it) {
        const int j0 = (threadIdx.x << 2) + (it << 10);
        float4 v;
        v.x = -fabsf(pi - pos[j0 + 0]);
        v.y = -fabsf(pi - pos[j0 + 1]);
        v.z = -fabsf(pi - pos[j0 + 2]);
        v.w = -fabsf(pi - pos[j0 + 3]);
        *(float4*)(row + j0) = v;
    }
}

// ---------------------------------------------------------------------------
// fp32 WMMA 16x16 tile core: per-wave compute over one staged 16x64 A panel.
//   A frag (16x4 f32): lane l (0-15): row=l,   V0/V1 = K {0,1}
//                      lane l (16-31): row=l-16, V0/V1 = K {2,3}
//   B frag (4x16 f32): lane l: col = l&15, K = {0,1} (+2 for upper half)
//   C (16x16 f32, 8 VGPRs): M = v + 8*(lane>=16), N = lane&15
// ---------------------------------------------------------------------------
__device__ __forceinline__ v8f wmma_panel(const float* a_sh, const float* wcol,
                                          int l15, int half, v8f acc)
{
#pragma unroll
    for (int t = 0; t < 16; ++t) {
        v2f a = *(const v2f*)(&a_sh[l15 * LDA_SH + (t << 2) + half * 2]);
        v2f b = *(const v2f*)(wcol + (t << 2));
        acc = __builtin_amdgcn_wmma_f32_16x16x4_f32(
            /*neg_a=*/false, a, /*neg_b=*/false, b,
            /*c_mod=*/(short)0, acc, /*reuse_a=*/false, /*reuse_b=*/false);
    }
    return acc;
}

__device__ __forceinline__ void wmma_epilogue(v8f acc, const float* __restrict__ bias,
                                              float* __restrict__ C, int ldc, int coloff,
                                              int row0, int col, int half, int applyGelu)
{
    const float bcol = bias[col];
#pragma unroll
    for (int v = 0; v < 8; ++v) {
        const int r = row0 + v + (half << 3);
        float val = acc[v] + bcol;
        if (applyGelu)
            val = 0.5f * val * (1.0f + erff(val * 0.70710678118654752f)); // exact GELU
        C[(size_t)r * ldc + coloff + col] = val;
    }
}

// ---------------------------------------------------------------------------
// Kernel 3: generic fp32 WMMA GEMM,  C[:, coloff:coloff+N] = A @ W^T + bias.
//   A row i: k < ksplit -> A0[i*lda0+k], else A1[i*lda1+(k-ksplit)]
//   (two-source A implements the concat([set_states, router_counts]) GEMM)
//   W is [N, K] row-major, so B[k,n] = W[n*K + k]  (contiguous in k).
// grid = (N/128, M/16); block = 256 (8 waves, one 16x16 tile each).
// ---------------------------------------------------------------------------
__global__ void wmma_gemm_f32(const float* __restrict__ A0, const float* __restrict__ A1,
                              int ksplit, int lda0, int lda1,
                              const float* __restrict__ W, int K,
                              const float* __restrict__ bias,
                              float* __restrict__ C, int ldc, int coloff,
                              int applyGelu)
{
    __shared__ __align__(16) float a_sh[16 * LDA_SH];
    const int wave = threadIdx.x >> 5;
    const int lane = threadIdx.x & 31;
    const int half = lane >> 4;
    const int l15  = lane & 15;
    const int row0 = blockIdx.y << 4;
    const int col  = (blockIdx.x << 7) + (wave << 4) + l15;

    v8f acc = {0.f, 0.f, 0.f, 0.f, 0.f, 0.f, 0.f, 0.f};

    for (int kk = 0; kk < K; kk += 64) {
        // Stage A panel [16 rows x 64 K] into LDS (coalesced, 4 elems/thread).
#pragma unroll
        for (int i = 0; i < 4; ++i) {
            const int f  = threadIdx.x + (i << 8);
            const int r  = f >> 6;
            const int k  = f & 63;
            const int kg = kk + k;
            const float v = (kg < ksplit)
                ? A0[(size_t)(row0 + r) * lda0 + kg]
                : A1[(size_t)(row0 + r) * lda1 + (kg - ksplit)];
            a_sh[r * LDA_SH + k] = v;
        }
        __syncthreads();
        const float* wcol = W + (size_t)col * K + kk + half * 2;
        acc = wmma_panel(a_sh, wcol, l15, half, acc);
        __syncthreads();
    }
    wmma_epilogue(acc, bias, C, ldc, coloff, row0, col, half, applyGelu);
}

// ---------------------------------------------------------------------------
// Kernel 4: geometric GEMM: C = exp(-|pos_i - pos_k|) @ geom_w^T + geom_b.
// A is computed analytically into LDS during staging -> the 64 MB geom_row
// matrix is never materialized. K = MSETS = 4096.
// ---------------------------------------------------------------------------
__global__ void wmma_gemm_geom(const float* __restrict__ pos,
                               const float* __restrict__ W,   // [D_PHI, MSETS]
                               const float* __restrict__ bias,
                               float* __restrict__ C, int ldc)
{
    __shared__ __align__(16) float a_sh[16 * LDA_SH];
    const int wave = threadIdx.x >> 5;
    const int lane = threadIdx.x & 31;
    const int half = lane >> 4;
    const int l15  = lane & 15;
    const int row0 = blockIdx.y << 4;
    const int col  = (blockIdx.x << 7) + (wave << 4) + l15;

    v8f acc = {0.f, 0.f, 0.f, 0.f, 0.f, 0.f, 0.f, 0.f};

    for (int kk = 0; kk < MSETS; kk += 64) {
#pragma unroll
        for (int i = 0; i < 4; ++i) {
            const int f  = threadIdx.x + (i << 8);
            const int r  = f >> 6;
            const int k  = f & 63;
            a_sh[r * LDA_SH + k] = expf(-fabsf(pos[row0 + r] - pos[kk + k]));
        }
        __syncthreads();
        const float* wcol = W + (size_t)col * MSETS + kk + half * 2;
        acc = wmma_panel(a_sh, wcol, l15, half, acc);
        __syncthreads();
    }
    wmma_epilogue(acc, bias, C, ldc, /*coloff=*/0, row0, col, half, /*gelu=*/0);
}

// ---------------------------------------------------------------------------
// Launch: inputs in setup_inputs() order; outputs concatenated
// (phi_attn [4096,512], desc_router [4096,1024], geom_bias [4096,4096]).
// ---------------------------------------------------------------------------
extern "C" void kernel_launch(void* const* d_in, const int* in_sizes, int n_in,
                              void* d_out, int out_size, void* d_ws, size_t ws_size,
                              hipStream_t stream)
{
    const int*   token_ids   = (const int*)  d_in[0];
    const int*   set_indices = (const int*)  d_in[1];
    const int*   set_sizes   = (const int*)  d_in[2];
    const float* set_pos     = (const float*)d_in[3];
    const float* set_states  = (const float*)d_in[4];
    const float* geom_w      = (const float*)d_in[5];
    const float* geom_b      = (const float*)d_in[6];
    const float* count_w     = (const float*)d_in[7];
    const float* count_b     = (const float*)d_in[8];
    const float* rc_w        = (const float*)d_in[9];
    const float* rc_b        = (const float*)d_in[10];
    const float* rf_w        = (const float*)d_in[11];
    const float* rf_b        = (const float*)d_in[12];
    const float* f1_w        = (const float*)d_in[13];
    const float* f1_b        = (const float*)d_in[14];
    const float* f2_w        = (const float*)d_in[15];
    const float* f2_b        = (const float*)d_in[16];

    float* out         = (float*)d_out;
    float* phi_attn    = out;                                         // 4096*512
    float* desc_router = out + (size_t)MSETS * D_PHI;                 // 4096*1024
    float* geom_bias   = desc_router + (size_t)MSETS * D_MODEL;       // 4096*4096

    float* ws     = (float*)d_ws;
    float* counts = ws;                                   // [4096, 128]
    float* fused  = counts + (size_t)MSETS * NUM_BINS;    // [4096, 1024]
    float* hbuf   = fused  + (size_t)MSETS * (2 * D_PHI); // [4096, 512]
    float* rcbuf  = hbuf   + (size_t)MSETS * D_PHI;       // [4096, 1024]

    const dim3 blk(256);
    const dim3 gN512(D_PHI / 128, MSETS / 16);    // (4, 256)
    const dim3 gN1024(D_MODEL / 128, MSETS / 16); // (8, 256)

    // 1) normalized hashed-count histograms
    counts_kernel<<<MSETS, blk, 0, stream>>>(token_ids, set_indices, set_sizes, counts);
    // 2) geom_bias output (independent of everything else)
    geom_bias_kernel<<<MSETS, blk, 0, stream>>>(set_pos, geom_bias);
    // 3) proj_geom = exp(geom_bias) @ geom_w.T + geom_b -> fused[:, 0:512]
    wmma_gemm_geom<<<gN512, blk, 0, stream>>>(set_pos, geom_w, geom_b, fused, 2 * D_PHI);
    // 4) proj_counts = counts @ count_w.T + count_b -> fused[:, 512:1024]
    wmma_gemm_f32<<<gN512, blk, 0, stream>>>(counts, counts, NUM_BINS, NUM_BINS, NUM_BINS,
                                             count_w, NUM_BINS, count_b,
                                             fused, 2 * D_PHI, D_PHI, 0);
    // 5) h = gelu(fused @ f1_w.T + f1_b)
    wmma_gemm_f32<<<gN512, blk, 0, stream>>>(fused, fused, 2 * D_PHI, 2 * D_PHI, 2 * D_PHI,
                                             f1_w, 2 * D_PHI, f1_b,
                                             hbuf, D_PHI, 0, 1);
    // 6) phi_attn = h @ f2_w.T + f2_b
    wmma_gemm_f32<<<gN512, blk, 0, stream>>>(hbuf, hbuf, D_PHI, D_PHI, D_PHI,
                                             f2_w, D_PHI, f2_b,
                                             phi_attn, D_PHI, 0, 0);
    // 7) router_counts = counts @ rc_w.T + rc_b
    wmma_gemm_f32<<<gN1024, blk, 0, stream>>>(counts, counts, NUM_BINS, NUM_BINS, NUM_BINS,
                                              rc_w, NUM_BINS, rc_b,
                                              rcbuf, D_MODEL, 0, 0);
    // 8) desc_router = concat([set_states, router_counts]) @ rf_w.T + rf_b
    //    (two-source A: k < 1024 -> set_states, else rcbuf)
    wmma_gemm_f32<<<gN1024, blk, 0, stream>>>(set_states, rcbuf, D_MODEL, D_MODEL, D_MODEL,
                                              rf_w, 2 * D_MODEL, rf_b,
                                              desc_router, D_MODEL, 0, 0);
}